// Wide_2087354106706
// MI455X (gfx1250) — compile-verified
//
#include <hip/hip_runtime.h>

#define IN_DIM      512
#define NUM_CLASSES 10
#define QUAD_DIM    (IN_DIM + IN_DIM * (IN_DIM + 1) / 2)   // 131840
#define BATCH       1024
#define NSLICES     64
#define KSLICE      (QUAD_DIM / NSLICES)                   // 2060 (multiple of 4)
#define NBT         (BATCH / 16)                           // 64
#define NPAIR       (BATCH / 32)                           // 32 tile-pairs
#define XS_STRIDE   516                                    // floats; 516%64==4 -> conflict-free, 16B-aligned rows
#define TILE_B_OFF  (16 * XS_STRIDE * 4)                   // 33024 B, fits 16-bit DS offset
#define TAB_BYTES   (QUAD_DIM * 4)                         // 527360, multiple of 256

typedef float v2f __attribute__((ext_vector_type(2)));
typedef float v8f __attribute__((ext_vector_type(8)));

__device__ __forceinline__ int tri_off(int i) {
    return i * IN_DIM - ((i * (i - 1)) >> 1);
}

// One-shot: tab[k] = packed LDS byte offsets ((j*4)<<16)|(i*4) such that
// x_tilde[:,k] == x[:,i]*x[:,j] with the convention x[:,512] == 1.0 (linear terms).
__global__ void build_table(unsigned* __restrict__ tab) {
    int k = blockIdx.x * 256 + threadIdx.x;
    if (k >= QUAD_DIM) return;
    int i, j;
    if (k < IN_DIM) {
        i = k; j = IN_DIM;                       // linear term: x[k] * 1.0
    } else {
        int t = k - IN_DIM;
        float r = sqrtf(fmaxf(1050625.0f - 8.0f * (float)t, 0.0f));  // (2p+1)^2 - 8t
        int ii = (int)((1025.0f - r) * 0.5f);
        ii = ii < 0 ? 0 : (ii > IN_DIM - 1 ? IN_DIM - 1 : ii);
        if (ii < IN_DIM - 1 && tri_off(ii + 1) <= t) ++ii;           // float fix-ups
        if (ii < IN_DIM - 1 && tri_off(ii + 1) <= t) ++ii;
        if (ii > 0 && tri_off(ii) > t) --ii;
        if (ii > 0 && tri_off(ii) > t) --ii;
        i = ii;
        j = ii + (t - tri_off(ii));
    }
    tab[k] = ((unsigned)(j * 4) << 16) | (unsigned)(i * 4);
}

// One wave = TWO 16x16 f32 accumulator tiles (32 batch rows x 16 class cols) sharing
// one B fragment and one index load per K-step; split-K over NSLICES slices.
__global__ __launch_bounds__(128) void quadfeat_gemm(const float* __restrict__ X,
                                                     const float* __restrict__ W,
                                                     const unsigned* __restrict__ tab,
                                                     float* __restrict__ partial) {
    __shared__ float xs[32 * XS_STRIDE];                   // ~66 KB: two 16x512 x-tiles

    const int pair = blockIdx.x;                           // 0..31 -> batch rows pair*32..
    const int tid  = threadIdx.x;

    // cooperative float4 fill of 32x512 rows (row starts 16B aligned)
    const float4* src4 = (const float4*)(X + (size_t)pair * (32 * IN_DIM));
    for (int v = tid; v < 4096; v += 128) {
        int row = v >> 7;                                  // 128 float4 per row
        int c4  = v & 127;
        *(float4*)(xs + row * XS_STRIDE + c4 * 4) = src4[v];
    }
    if (tid < 32) xs[tid * XS_STRIDE + IN_DIM] = 1.0f;     // constant-1 slot per row
    __syncthreads();

    const int wave  = tid >> 5;                            // wave32
    const int lane  = tid & 31;
    const int slice = blockIdx.y * 4 + wave;               // 0..63
    const int hi    = lane >> 4;
    const int n     = lane & 15;                           // class col AND batch row (ISA layout)
    const char* xrow = (const char*)(xs + n * XS_STRIDE);  // tile A row; tile B = +TILE_B_OFF

    const int k0 = slice * KSLICE + 2 * hi;                // this lane's K elements: k0, k0+1
    const unsigned* tp = tab + k0;
    // padded classes 10..15 read row 9; their output columns are never consumed
    const float* wp = W + (size_t)((n < NUM_CLASSES) ? n : (NUM_CLASSES - 1)) * QUAD_DIM + k0;

    v8f accA = {0.f, 0.f, 0.f, 0.f, 0.f, 0.f, 0.f, 0.f};
    v8f accB = {0.f, 0.f, 0.f, 0.f, 0.f, 0.f, 0.f, 0.f};

    #pragma unroll 2
    for (int it = 0; it < KSLICE / 4; ++it) {
        uint2 idx = *(const uint2*)tp;                     // packed byte offsets for k, k+1
        unsigned o_i0 = idx.x & 0xFFFFu, o_j0 = idx.x >> 16;
        unsigned o_i1 = idx.y & 0xFFFFu, o_j1 = idx.y >> 16;

        // tile A (rows 0..15): 4 ds loads
        float ai0 = *(const float*)(xrow + o_i0);
        float aj0 = *(const float*)(xrow + o_j0);
        float ai1 = *(const float*)(xrow + o_i1);
        float aj1 = *(const float*)(xrow + o_j1);
        // tile B (rows 16..31): same vaddr + constant DS offset
        float bi0 = *(const float*)(xrow + TILE_B_OFF + o_i0);
        float bj0 = *(const float*)(xrow + TILE_B_OFF + o_j0);
        float bi1 = *(const float*)(xrow + TILE_B_OFF + o_i1);
        float bj1 = *(const float*)(xrow + TILE_B_OFF + o_j1);

        v2f b = *(const v2f*)wp;                           // shared B: W^T chunk

        v2f aA; aA.x = ai0 * aj0; aA.y = ai1 * aj1;
        v2f aB; aB.x = bi0 * bj0; aB.y = bi1 * bj1;

        accA = __builtin_amdgcn_wmma_f32_16x16x4_f32(
            false, aA, false, b, (short)0, accA, false, false);
        accB = __builtin_amdgcn_wmma_f32_16x16x4_f32(
            false, aB, false, b, (short)0, accB, false, false);

        tp += 4;
        wp += 4;
    }

    // store both partial tiles: C layout VGPR r -> M = r + 8*hi, N = lane&15
    const int bt0 = pair * 2;
    float* pA = partial + ((size_t)slice * NBT + bt0) * 256;
    float* pB = partial + ((size_t)slice * NBT + bt0 + 1) * 256;
    #pragma unroll
    for (int r = 0; r < 8; ++r) {
        pA[(r + 8 * hi) * 16 + n] = accA[r];
        pB[(r + 8 * hi) * 16 + n] = accB[r];
    }
}

// deterministic fixed-order split-K reduction + bias; writes every d_out element
__global__ void quadfeat_reduce(const float* __restrict__ partial,
                                const float* __restrict__ bias,
                                float* __restrict__ out) {
    int idx = blockIdx.x * blockDim.x + threadIdx.x;
    if (idx >= BATCH * NUM_CLASSES) return;
    int b = idx / NUM_CLASSES;
    int c = idx - b * NUM_CLASSES;
    int bt = b >> 4;
    int m  = b & 15;
    float s = bias[c];
    #pragma unroll 4
    for (int sl = 0; sl < NSLICES; ++sl)
        s += partial[((size_t)sl * NBT + bt) * 256 + m * 16 + c];
    out[(size_t)b * NUM_CLASSES + c] = s;
}

extern "C" void kernel_launch(void* const* d_in, const int* in_sizes, int n_in,
                              void* d_out, int out_size, void* d_ws, size_t ws_size,
                              hipStream_t stream) {
    const float* x    = (const float*)d_in[0];   // [1024, 512]
    const float* W    = (const float*)d_in[1];   // [10, 131840]
    const float* bias = (const float*)d_in[2];   // [10]
    float* out        = (float*)d_out;           // [1024, 10]

    unsigned* tab  = (unsigned*)d_ws;                          // 527 360 B
    float* partial = (float*)((char*)d_ws + TAB_BYTES);        // 4 MB

    build_table<<<(QUAD_DIM + 255) / 256, 256, 0, stream>>>(tab);

    dim3 grid(NPAIR, NSLICES / 4);               // 32 x 16 blocks, 128 threads (4 waves)
    quadfeat_gemm<<<grid, 128, 0, stream>>>(x, W, tab, partial);

    int total = BATCH * NUM_CLASSES;
    quadfeat_reduce<<<(total + 255) / 256, 256, 0, stream>>>(partial, bias, out);
}